// SmileResampler_5145370821359
// MI455X (gfx1250) — compile-verified
//
#include <hip/hip_runtime.h>
#include <cstdint>

// SmileResampler for MI455X (gfx1250).
// Memory-bound band-lerp gather: ~270 MB HBM traffic -> ~11.5 us at 23.3 TB/s.
// WMMA is inapplicable (2-nnz/row band interp as dense matmul would be ~250x
// more work). We instead use the CDNA5 async global->LDS path:
//   global_load_async_to_lds_b128 + s_wait_asynccnt double buffering,
// turning the band-axis gather into conflict-free LDS reads (row stride = 64
// floats = 64 banks -> bank = w%64 independent of gathered band).

namespace {
constexpr int kB  = 4;
constexpr int kBh = 128;
constexpr int kH  = 256;
constexpr int kW  = 256;

constexpr int WT  = 64;                 // w-tile per block
constexpr int HB  = 8;                  // h rows per block
constexpr int NT  = 256;                // threads per block (8 waves of 32)
constexpr int TILE   = kBh * WT;        // floats per x tile   (8192 = 32 KB)
constexpr int CHUNKS = (TILE * 4) / 16; // b128 chunks per tile (2048)
constexpr int CPT    = CHUNKS / NT;     // chunks per thread    (8)
}

__device__ __forceinline__ void async_ld_b128(uint32_t lds_addr, int goff,
                                              const void* sbase) {
  // GVS mode: mem = SGPR64 base + signed VGPR32 offset; LDS dest addr in VGPR.
  asm volatile("global_load_async_to_lds_b128 %0, %1, %2"
               :
               : "v"(lds_addr), "v"(goff), "s"(sbase)
               : "memory");
}

__device__ __forceinline__ void issue_tile(const float* __restrict__ x, int b,
                                           int h, int w0, uint32_t lds_base,
                                           int t) {
#pragma unroll
  for (int i = 0; i < CPT; ++i) {
    int c    = i * NT + t;       // 0..2047
    int band = c >> 4;           // 16 x b128 chunks per 64-float band row
    int wc   = c & 15;
    int goff = (((b * kBh + band) * kH + h) * kW + w0) * 4 + wc * 16;
    uint32_t lds = lds_base + (uint32_t)(band * (WT * 4) + wc * 16);
    async_ld_b128(lds, goff, x);
  }
}

__global__ __launch_bounds__(NT) void smile_resample_kernel(
    const float* __restrict__ x, const float* __restrict__ wsh,
    float* __restrict__ out) {
  __shared__ uint2 meta[TILE];        // 64 KB: {off0|off1<<16, bits(wfrac)}
  __shared__ float xtile[2][TILE];    // 2 x 32 KB double buffer

  const int t   = threadIdx.x;
  const int blk = blockIdx.x;
  const int nw  = kW / WT;            // 4
  const int nh  = kH / HB;            // 32
  const int b   = blk / (nw * nh);
  const int r   = blk - b * (nw * nh);
  const int wt  = r / nh;
  const int ht  = r - wt * nh;
  const int w0  = wt * WT;
  const int h0  = ht * HB;

  const uint32_t xb0 = (uint32_t)(uintptr_t)(void*)&xtile[0][0];
  const uint32_t xb1 = (uint32_t)(uintptr_t)(void*)&xtile[1][0];

  // Kick off the first tile load, then overlap the per-block weight precompute.
  issue_tile(x, b, h0, w0, xb0, t);   // ASYNCcnt = 8 per wave

  // h-invariant weights/indices: compute once per block, reuse for HB rows.
  const float scale = (float)(128.0 / 127.0);  // f32-rounded Bh/(Bh-1)
#pragma unroll 4
  for (int e = t; e < TILE; e += NT) {
    int   band = e >> 6;             // e / WT
    int   wl   = e & (WT - 1);
    float v    = wsh[(b * kBh + band) * kW + w0 + wl];
    float s    = fminf(fmaxf(v, -2.0f), 2.0f);
    float sh   = fminf(fmaxf((float)band + s, 0.0f), 127.0f);
    float iy   = sh * scale - 0.5f;
    float fl   = floorf(iy);
    float wf   = iy - fl;
    int   i0   = (int)fl;
    int   i0c  = i0 < 0 ? 0 : (i0 > kBh - 1 ? kBh - 1 : i0);
    int   i1   = i0 + 1;
    int   i1c  = i1 < 0 ? 0 : (i1 > kBh - 1 ? kBh - 1 : i1);
    uint2 m;
    m.x = (uint32_t)(i0c * WT + wl) | ((uint32_t)(i1c * WT + wl) << 16);
    m.y = __float_as_uint(wf);
    meta[e] = m;
  }

  const int wl = t & (WT - 1);
  const int bg = t >> 6;             // 0..3: band group within an iteration
  int sel = 0;

  for (int hi = 0; hi < HB; ++hi) {
    const int h = h0 + hi;
    if (hi + 1 < HB) {
      // Prefetch next row's tile into the other buffer, then retire current.
      issue_tile(x, b, h + 1, w0, sel ? xb0 : xb1, t);
      asm volatile("s_wait_asynccnt 0x8" ::: "memory");
    } else {
      asm volatile("s_wait_asynccnt 0x0" ::: "memory");
    }
    __syncthreads();  // all waves' async loads into current buffer visible

    const float* __restrict__ buf = xtile[sel];
    const unsigned obase =
        ((unsigned)(b * kBh) * kH + (unsigned)h) * kW + (unsigned)(w0 + wl);
#pragma unroll 4
    for (int k = 0; k < kBh / 4; ++k) {
      int   band = (k << 2) | bg;
      uint2 m    = meta[band * WT + wl];          // ds_load_b64
      float wf   = __uint_as_float(m.y);
      float a    = buf[m.x & 0xFFFFu];            // ds_load_b32, bank = wl
      float c    = buf[m.x >> 16];                // ds_load_b32, bank = wl
      out[obase + (unsigned)band * (unsigned)(kH * kW)] =
          a * (1.0f - wf) + c * wf;
    }
    __syncthreads();  // everyone done reading buf before it is overwritten
    sel ^= 1;
  }
}

extern "C" void kernel_launch(void* const* d_in, const int* in_sizes, int n_in,
                              void* d_out, int out_size, void* d_ws,
                              size_t ws_size, hipStream_t stream) {
  (void)in_sizes; (void)n_in; (void)d_ws; (void)ws_size; (void)out_size;
  const float* x   = (const float*)d_in[0];
  const float* wsh = (const float*)d_in[1];
  float*       out = (float*)d_out;

  const int blocks = kB * (kW / WT) * (kH / HB);  // 512
  smile_resample_kernel<<<dim3(blocks), dim3(NT), 0, stream>>>(x, wsh, out);
}